// UniGATConv_9912784519287
// MI455X (gfx1250) — compile-verified
//
#include <hip/hip_runtime.h>
#include <math.h>

typedef __attribute__((ext_vector_type(2))) float v2f;
typedef __attribute__((ext_vector_type(8))) float v8f;

#define HEADS    4
#define DHEAD    64
#define CHANNELS 256   // HEADS * DHEAD
#define INSZ     256
#define NEDGES   10000

// ---------------------------------------------------------------------------
// Workspace init: he=0, deg=0, alpha=0, mx=-inf, den=0, out=0
// ---------------------------------------------------------------------------
__global__ void init_ws_kernel(float* he, float* deg, float* alpha,
                               float* mx, float* den, float* out,
                               int E, int N) {
  const long long i = (long long)blockIdx.x * blockDim.x + threadIdx.x;
  if (i < (long long)E * CHANNELS) he[i] = 0.f;
  if (i < E)                       deg[i] = 0.f;
  if (i < (long long)E * HEADS)    alpha[i] = 0.f;
  if (i < (long long)N * HEADS)  { mx[i] = -INFINITY; den[i] = 0.f; }
  if (i < (long long)N * CHANNELS) out[i] = 0.f;
}

// ---------------------------------------------------------------------------
// xp = x @ W  via V_WMMA_F32_16X16X4_F32 (fp32 matrix core, exact precision)
// One wave computes a 16(M) x 64(N) strip: 4 accumulators share the A fragment.
// Layouts per CDNA5 ISA 7.12.2:
//   A 16x4 f32 : lanes 0-15 -> M=lane, v0=K0 v1=K1 ; lanes 16-31 -> v0=K2 v1=K3
//   B 4x16 f32 : lanes 0-15 -> N=lane, v0=K0 v1=K1 ; lanes 16-31 -> v0=K2 v1=K3
//   C/D 16x16  : VGPR r, lanes 0-15 -> (M=r, N=lane); lanes 16-31 -> (M=r+8)
// ---------------------------------------------------------------------------
__global__ __launch_bounds__(256)
void gemm_xw_wmma(const float* __restrict__ x, const float* __restrict__ W,
                  float* __restrict__ xp, int N) {
  const int wave = blockIdx.x * 8 + (threadIdx.x >> 5);
  const int lane = threadIdx.x & 31;
  const int col_strips = CHANNELS / 64;          // 4 strips of 64 cols
  const int mt = wave / col_strips;
  const int m0 = mt * 16;
  if (m0 >= N) return;
  const int n0  = (wave % col_strips) * 64;
  const int l15 = lane & 15;
  const int lh  = lane >> 4;                     // K half select

  v8f acc[4] = {};                               // zero accumulators

  const int mrow = (m0 + l15 < N) ? (m0 + l15) : (N - 1);
  const float* xr = x + (size_t)mrow * INSZ;

  for (int kb = 0; kb < INSZ; kb += 4) {
    const int ka = kb + lh * 2;
    v2f a;
    a.x = xr[ka];
    a.y = xr[ka + 1];
    const float* w0 = W + (size_t)ka * CHANNELS + n0 + l15;
    const float* w1 = w0 + CHANNELS;
#pragma unroll
    for (int t = 0; t < 4; ++t) {
      v2f b;
      b.x = w0[t * 16];
      b.y = w1[t * 16];
      acc[t] = __builtin_amdgcn_wmma_f32_16x16x4_f32(
          /*neg_a=*/false, a, /*neg_b=*/false, b,
          /*c_mod=*/(short)0, acc[t], /*reuse_a=*/false, /*reuse_b=*/false);
    }
  }

#pragma unroll
  for (int t = 0; t < 4; ++t) {
#pragma unroll
    for (int r = 0; r < 8; ++r) {
      const int m = m0 + lh * 8 + r;
      if (m < N) xp[(size_t)m * CHANNELS + n0 + t * 16 + l15] = acc[t][r];
    }
  }
}

// ---------------------------------------------------------------------------
// Stage 1: he[e] += xp[node]  (64 lanes x float4 per membership) + degree
// ---------------------------------------------------------------------------
__global__ __launch_bounds__(256)
void scatter_he_kernel(const float* __restrict__ xp,
                       const int* __restrict__ node_idx,
                       const int* __restrict__ hedge_idx,
                       float* he, float* deg, int M) {
  const int t = threadIdx.x;
  const int memb = blockIdx.x * 4 + (t >> 6);
  if (memb >= M) return;
  const int c0 = (t & 63) * 4;
  const int v = node_idx[memb];
  const int e = hedge_idx[memb];
  const float4 val = *(const float4*)(xp + (size_t)v * CHANNELS + c0);
  float* dst = he + (size_t)e * CHANNELS + c0;
  atomicAdd(dst + 0, val.x);
  atomicAdd(dst + 1, val.y);
  atomicAdd(dst + 2, val.z);
  atomicAdd(dst + 3, val.w);
  if ((t & 63) == 0) atomicAdd(deg + e, 1.0f);
}

// ---------------------------------------------------------------------------
// Normalize he by degree (in place) and compute alpha_e = lrelu(<he, att_e>)
// One wave32 per (edge, head).
// ---------------------------------------------------------------------------
__global__ __launch_bounds__(256)
void edge_alpha_kernel(float* he, const float* __restrict__ deg,
                       const float* __restrict__ att, float* alpha, int E) {
  const int wid  = blockIdx.x * 8 + (threadIdx.x >> 5);
  const int lane = threadIdx.x & 31;
  const int e = wid >> 2;
  const int h = wid & 3;
  if (e >= E) return;
  const float inv = 1.0f / fmaxf(deg[e], 1.0f);
  float* hep = he + (size_t)e * CHANNELS + h * DHEAD;
  const float* ap = att + h * DHEAD;
  float v0 = hep[lane] * inv;
  float v1 = hep[lane + 32] * inv;
  hep[lane] = v0;
  hep[lane + 32] = v1;
  float s = v0 * ap[lane] + v1 * ap[lane + 32];
#pragma unroll
  for (int off = 16; off > 0; off >>= 1) s += __shfl_xor(s, off, 32);
  if (lane == 0) alpha[e * HEADS + h] = (s > 0.f) ? s : 0.2f * s;
}

// ---------------------------------------------------------------------------
// Edge-softmax pass 1: per-node max (order-independent int-bit atomic trick)
// ---------------------------------------------------------------------------
__global__ void scatter_max_kernel(const float* __restrict__ alpha,
                                   const int* __restrict__ node_idx,
                                   const int* __restrict__ hedge_idx,
                                   float* mx, int M) {
  const long long i = (long long)blockIdx.x * blockDim.x + threadIdx.x;
  if (i >= (long long)M * HEADS) return;
  const int memb = (int)(i >> 2);
  const int h    = (int)(i & 3);
  const float a = alpha[hedge_idx[memb] * HEADS + h];
  float* addr = mx + (size_t)node_idx[memb] * HEADS + h;
  if (a >= 0.f) atomicMax((int*)addr, __float_as_int(a));
  else          atomicMin((unsigned int*)addr, __float_as_uint(a));
}

// ---------------------------------------------------------------------------
// Edge-softmax pass 2: denominator
// ---------------------------------------------------------------------------
__global__ void scatter_den_kernel(const float* __restrict__ alpha,
                                   const int* __restrict__ node_idx,
                                   const int* __restrict__ hedge_idx,
                                   const float* __restrict__ mx,
                                   float* den, int M) {
  const long long i = (long long)blockIdx.x * blockDim.x + threadIdx.x;
  if (i >= (long long)M * HEADS) return;
  const int memb = (int)(i >> 2);
  const int h    = (int)(i & 3);
  const int v = node_idx[memb];
  const float a = alpha[hedge_idx[memb] * HEADS + h];
  atomicAdd(den + (size_t)v * HEADS + h, __expf(a - mx[(size_t)v * HEADS + h]));
}

// ---------------------------------------------------------------------------
// Stage 2: out[node] += a * he[edge]   (256 threads per membership)
// ---------------------------------------------------------------------------
__global__ __launch_bounds__(256)
void scatter_out_kernel(const float* __restrict__ he,
                        const float* __restrict__ alpha,
                        const int* __restrict__ node_idx,
                        const int* __restrict__ hedge_idx,
                        const float* __restrict__ mx,
                        const float* __restrict__ den,
                        float* out, int M) {
  const int memb = blockIdx.x;
  if (memb >= M) return;
  const int c = threadIdx.x;           // 0..255
  const int h = c >> 6;
  const int v = node_idx[memb];
  const int e = hedge_idx[memb];
  const float a = __expf(alpha[e * HEADS + h] - mx[(size_t)v * HEADS + h])
                  / den[(size_t)v * HEADS + h];
  atomicAdd(out + (size_t)v * CHANNELS + c, he[(size_t)e * CHANNELS + c] * a);
}

// ---------------------------------------------------------------------------
extern "C" void kernel_launch(void* const* d_in, const int* in_sizes, int n_in,
                              void* d_out, int out_size, void* d_ws, size_t ws_size,
                              hipStream_t stream) {
  const float* x         = (const float*)d_in[0];
  const float* W         = (const float*)d_in[1];
  const float* att       = (const float*)d_in[2];
  const int*   node_idx  = (const int*)d_in[3];
  const int*   hedge_idx = (const int*)d_in[4];
  const int N = in_sizes[0] / INSZ;
  const int M = in_sizes[3];
  const int E = NEDGES;
  float* out = (float*)d_out;

  // Workspace carve-up (floats): xp | he | deg | alpha | mx | den  (~63 MB)
  float* xp    = (float*)d_ws;
  float* he    = xp    + (size_t)N * CHANNELS;
  float* deg   = he    + (size_t)E * CHANNELS;
  float* alpha = deg   + E;
  float* mx    = alpha + (size_t)E * HEADS;
  float* den   = mx    + (size_t)N * HEADS;

  {
    long long total = (long long)N * CHANNELS;
    int blocks = (int)((total + 255) / 256);
    init_ws_kernel<<<blocks, 256, 0, stream>>>(he, deg, alpha, mx, den, out, E, N);
  }
  {
    int mtiles = (N + 15) / 16;
    int waves  = mtiles * (CHANNELS / 64);
    int blocks = (waves + 7) / 8;
    gemm_xw_wmma<<<blocks, 256, 0, stream>>>(x, W, xp, N);
  }
  {
    int blocks = (M + 3) / 4;
    scatter_he_kernel<<<blocks, 256, 0, stream>>>(xp, node_idx, hedge_idx, he, deg, M);
  }
  {
    int waves  = E * HEADS;
    int blocks = (waves + 7) / 8;
    edge_alpha_kernel<<<blocks, 256, 0, stream>>>(he, deg, att, alpha, E);
  }
  {
    long long total = (long long)M * HEADS;
    int blocks = (int)((total + 255) / 256);
    scatter_max_kernel<<<blocks, 256, 0, stream>>>(alpha, node_idx, hedge_idx, mx, M);
    scatter_den_kernel<<<blocks, 256, 0, stream>>>(alpha, node_idx, hedge_idx, mx, den, M);
  }
  scatter_out_kernel<<<M, 256, 0, stream>>>(he, alpha, node_idx, hedge_idx, mx, den, out, M);
}